// MCEwTAAModel_66529043415491
// MI455X (gfx1250) — compile-verified
//
#include <hip/hip_runtime.h>

// Problem constants (from reference)
#define V_SZ    50000
#define E_SZ    256
#define R_SZ    180
#define TW      (2 * R_SZ + 1)   // 361
#define B_SZ    2048
#define L_SZ    200
#define NOPT_SZ 10

typedef __attribute__((ext_vector_type(2))) float v2f;
typedef __attribute__((ext_vector_type(8))) float v8f;

// ---------------------------------------------------------------------------
// Kernel 1: per-vocab-row softmax statistics over the (V x 361) taa table.
// One wave32 per row; single pass (row cached in 12 regs/lane), shuffle-reduce.
// taa_bias is a per-row constant -> softmax(x + c) == softmax(x): it cancels.
// ---------------------------------------------------------------------------
__global__ __launch_bounds__(256) void taa_row_stats(const float* __restrict__ taa,
                                                     float2* __restrict__ stats) {
  const int row  = blockIdx.x * 8 + (threadIdx.x >> 5);
  const int lane = threadIdx.x & 31;
  if (row >= V_SZ) return;  // uniform per wave

  const float* r = taa + (size_t)row * TW;
  float vals[12];
  int cnt = 0;
  float m = -3.4e38f;
  for (int j = lane; j < TW; j += 32) {
    float v = r[j];
    vals[cnt++] = v;
    m = fmaxf(m, v);
  }
#pragma unroll
  for (int o = 16; o > 0; o >>= 1) m = fmaxf(m, __shfl_xor(m, o, 32));

  float s = 0.f;
  for (int i = 0; i < cnt; ++i) s += __expf(vals[i] - m);
#pragma unroll
  for (int o = 16; o > 0; o >>= 1) s += __shfl_xor(s, o, 32);

  if (lane == 0) stats[row] = make_float2(m, 1.0f / s);
}

// ---------------------------------------------------------------------------
// Kernel 2: one 256-thread block (8 waves) per batch element.
//   Phase A: 200 softmax weights from single-element taa gathers + row stats.
//   Phase B: hidden[e] = sum_l w[l] * ctx_emb[ev[l]][e].
//            4 groups x 64 threads; each thread owns 4 columns (float4/b128
//            loads, rows are 1KB-aligned); partials combined through LDS.
//   Phase C: scores(16x1 padded) = TE(16x256) x hidden(256) via 64 steps of
//            V_WMMA_F32_16X16X4_F32 on wave 0 (B operand N-replicated).
// ---------------------------------------------------------------------------
__global__ __launch_bounds__(256) void mce_taa_main(
    const int*   __restrict__ context,       // (B, L, 2)  [event, time]
    const int*   __restrict__ mixed_target,  // (B, 11)    [10 targets, 1 time]
    const float* __restrict__ ctx_emb,       // (V, 256)
    const float* __restrict__ tgt_emb,       // (V, 256)
    const float* __restrict__ taa,           // (V, 361)
    const float2* __restrict__ stats,        // (V)  {rowmax, 1/sumexp}
    float* __restrict__ out)                 // (B, 10)
{
  __shared__ float s_w[L_SZ];
  __shared__ int   s_ev[L_SZ];
  __shared__ float s_part[4][E_SZ];   // per-group float4 partials
  __shared__ float s_hidden[E_SZ];
  __shared__ int   s_tev[16];

  const int b   = blockIdx.x;
  const int tid = threadIdx.x;

  // uniform address -> scalar load
  const int ttime = mixed_target[b * (NOPT_SZ + 1) + NOPT_SZ];

  if (tid < L_SZ) {
    int ev = context[((size_t)b * L_SZ + tid) * 2 + 0];
    int tm = context[((size_t)b * L_SZ + tid) * 2 + 1];
    float msk = 1.0f;
    if (ev == -1) { ev = 0; msk = 0.0f; }
    if (ev < 0) ev = 0;
    if (ev >= V_SZ) ev = V_SZ - 1;
    int td = tm - ttime + R_SZ;
    td = td < 0 ? 0 : (td > 2 * R_SZ ? 2 * R_SZ : td);
    const float  x  = taa[(size_t)ev * TW + td];
    const float2 st = stats[ev];
    s_w[tid]  = __expf(x - st.x) * st.y * msk;
    s_ev[tid] = ev;
  }
  if (tid < 16) {
    int te = (tid < NOPT_SZ) ? mixed_target[b * (NOPT_SZ + 1) + tid] : 0;
    if (te < 0) te = 0;
    if (te >= V_SZ) te = V_SZ - 1;
    s_tev[tid] = te;
  }
  __syncthreads();

  // ---- Phase B: group g = tid/64 handles l = g, g+4, ... ; 4 cols/thread ----
  {
    const int g  = tid >> 6;        // 0..3
    const int c4 = tid & 63;        // float4 column index: cols 4*c4 .. 4*c4+3
    float4 acc = make_float4(0.f, 0.f, 0.f, 0.f);
#pragma unroll 5
    for (int l = g; l < L_SZ; l += 4) {
      const float wl = s_w[l];
      const float4 v = ((const float4*)(ctx_emb + (size_t)s_ev[l] * E_SZ))[c4];
      acc.x = fmaf(wl, v.x, acc.x);
      acc.y = fmaf(wl, v.y, acc.y);
      acc.z = fmaf(wl, v.z, acc.z);
      acc.w = fmaf(wl, v.w, acc.w);
    }
    *(float4*)&s_part[g][4 * c4] = acc;   // 16B-aligned LDS store
  }
  __syncthreads();

  // combine the 4 group partials (conflict-free: consecutive addresses)
  s_hidden[tid] = (s_part[0][tid] + s_part[1][tid]) +
                  (s_part[2][tid] + s_part[3][tid]);
  __syncthreads();

  // ---- Phase C: wave 0, scores via f32 WMMA 16x16x4 ----
  if (tid < 32) {
    const int lane  = tid;
    const int m     = lane & 15;       // target row (A matrix M index)
    const int khalf = lane >> 4;       // 0 -> K{0,1}, 1 -> K{2,3}
    const float* trow = tgt_emb + (size_t)s_tev[m] * E_SZ + 2 * khalf;

    v8f c = {};
#pragma unroll 4
    for (int kb = 0; kb < E_SZ / 4; ++kb) {
      const int base = kb * 4 + 2 * khalf;
      v2f a;  a.x  = trow[kb * 4];        a.y  = trow[kb * 4 + 1];
      v2f bb; bb.x = s_hidden[base];      bb.y = s_hidden[base + 1];  // N-replicated
      c = __builtin_amdgcn_wmma_f32_16x16x4_f32(
              /*neg_a=*/false, a, /*neg_b=*/false, bb,
              /*c_mod=*/(short)0, c, /*reuse_a=*/false, /*reuse_b=*/false);
    }

    // C layout: VGPR r, lanes 0-15 -> M=r ; lanes 16-31 -> M=8+r (all N equal)
    float* ob = out + (size_t)b * NOPT_SZ;
    if (lane == 0) {
#pragma unroll
      for (int r = 0; r < 8; ++r) ob[r] = c[r];
    } else if (lane == 16) {
      ob[8] = c[0];
      ob[9] = c[1];
    }
  }
}

// ---------------------------------------------------------------------------
extern "C" void kernel_launch(void* const* d_in, const int* in_sizes, int n_in,
                              void* d_out, int out_size, void* d_ws, size_t ws_size,
                              hipStream_t stream) {
  (void)in_sizes; (void)n_in; (void)out_size; (void)ws_size;

  const int*   context      = (const int*)d_in[0];
  const int*   mixed_target = (const int*)d_in[1];
  const float* ctx_emb      = (const float*)d_in[2];
  const float* tgt_emb      = (const float*)d_in[3];
  const float* taa          = (const float*)d_in[4];
  // d_in[5] (taa_bias) unused: per-row constant cancels inside softmax.

  float2* stats = (float2*)d_ws;  // 50000 * 8 B = 400 KB scratch, rewritten each call

  taa_row_stats<<<(V_SZ + 7) / 8, 256, 0, stream>>>(taa, stats);
  mce_taa_main<<<B_SZ, 256, 0, stream>>>(context, mixed_target, ctx_emb, tgt_emb,
                                         taa, stats, (float*)d_out);
}